// UnifiedFeatureWiseOCSVM_335007449267
// MI455X (gfx1250) — compile-verified
//
#include <hip/hip_runtime.h>
#include <stdint.h>

#define WSZ 5            // W_SIZE (compile-time, fully unrolled)
#define TPB 256          // threads per block = 8 wave32
#define FPT 4            // features per thread (float4-wide streams)
#define FPB (TPB * FPT)  // 1024 features per block
#define BPB 8            // batch rows per block (amortize W staging)

// ---- CDNA5 async global->LDS staging (ASYNCcnt path) ----
__device__ __forceinline__ uint32_t lds_off_of(const void* p) {
    // generic pointer to LDS: low 32 bits are the wave-relative LDS byte offset
    return (uint32_t)(uintptr_t)p;
}
__device__ __forceinline__ void async_to_lds_b128(uint32_t lds_off, const void* gaddr) {
    asm volatile("global_load_async_to_lds_b128 %0, %1, off"
                 :: "v"(lds_off), "v"(gaddr) : "memory");
}
__device__ __forceinline__ void async_to_lds_b32(uint32_t lds_off, const void* gaddr) {
    asm volatile("global_load_async_to_lds_b32 %0, %1, off"
                 :: "v"(lds_off), "v"(gaddr) : "memory");
}
__device__ __forceinline__ void wait_async0() {
    asm volatile("s_wait_asynccnt 0" ::: "memory");
}

__global__ __launch_bounds__(TPB)
void ocsvm_score_kernel(const float* __restrict__ x, const float* __restrict__ W,
                        const float* __restrict__ bias, float* __restrict__ out,
                        int B, int F) {
    __shared__ float wsh[FPB * WSZ];  // 20 KB: W[fbase..fbase+1023][0..4]
    __shared__ float bsh[FPB];        //  4 KB: bias chunk

    const int tid   = threadIdx.x;
    const int fbase = blockIdx.x * FPB;
    const int validF = min(FPB, F - fbase);
    if (validF <= 0) return;

    // ---- stage W chunk (contiguous validF*5 floats) into LDS, async ----
    {
        const int   wBytes = validF * WSZ * 4;
        const int   n128   = wBytes >> 4;
        const char* gsrc   = (const char*)(W + (size_t)fbase * WSZ);
        const uint32_t base = lds_off_of(wsh);
        for (int i = tid; i < n128; i += TPB)
            async_to_lds_b128(base + (uint32_t)i * 16u, gsrc + (size_t)i * 16);
        const int rem = (wBytes & 15) >> 2;  // 4-byte tail words
        if (tid < rem)
            async_to_lds_b32(base + (uint32_t)(n128 * 16 + tid * 4),
                             gsrc + (size_t)n128 * 16 + (size_t)tid * 4);
    }
    // ---- stage bias chunk into LDS, async ----
    {
        const int   bBytes = validF * 4;
        const int   n128   = bBytes >> 4;
        const char* gsrc   = (const char*)(bias + fbase);
        const uint32_t base = lds_off_of(bsh);
        for (int i = tid; i < n128; i += TPB)
            async_to_lds_b128(base + (uint32_t)i * 16u, gsrc + (size_t)i * 16);
        const int rem = (bBytes & 15) >> 2;
        if (tid < rem)
            async_to_lds_b32(base + (uint32_t)(n128 * 16 + tid * 4),
                             gsrc + (size_t)n128 * 16 + (size_t)tid * 4);
    }
    wait_async0();     // this wave's async copies landed in LDS
    __syncthreads();   // all waves' copies visible

    // ---- pull this thread's 20 weights + 4 biases into registers ----
    const int fl = tid * FPT;
    float wr[FPT][WSZ];
    float br[FPT];
#pragma unroll
    for (int j = 0; j < FPT; ++j) {
        br[j] = bsh[fl + j];
#pragma unroll
        for (int w = 0; w < WSZ; ++w)
            wr[j][w] = wsh[(fl + j) * WSZ + w];
    }

    const int  f0   = fbase + fl;
    const bool full = (fl + FPT) <= validF;

    const int bStart = blockIdx.y * BPB;
    const int bEnd   = min(B, bStart + BPB);

    for (int b = bStart; b < bEnd; ++b) {
        const size_t xoff = ((size_t)b * WSZ) * (size_t)F + (size_t)f0;
        if (full) {
            float4 xv[WSZ];
#pragma unroll
            for (int w = 0; w < WSZ; ++w)
                xv[w] = *(const float4*)(x + xoff + (size_t)w * F);

            // keep next batch's five 100KB-strided planes in flight
            if (b + 1 < bEnd) {
#pragma unroll
                for (int w = 0; w < WSZ; ++w)
                    __builtin_prefetch(x + xoff + (size_t)(WSZ + w) * F, 0, 1);
            }

            float4 o;
            float* op = (float*)&o;
#pragma unroll
            for (int j = 0; j < FPT; ++j) {
                float acc = br[j];
#pragma unroll
                for (int w = 0; w < WSZ; ++w)
                    acc = fmaf(((const float*)&xv[w])[j], wr[j][w], acc);
                op[j] = -acc;
            }
            *(float4*)(out + (size_t)b * F + f0) = o;
        } else {
#pragma unroll
            for (int j = 0; j < FPT; ++j) {
                if (fl + j < validF) {
                    float acc = br[j];
#pragma unroll
                    for (int w = 0; w < WSZ; ++w)
                        acc = fmaf(x[xoff + (size_t)w * F + j], wr[j][w], acc);
                    out[(size_t)b * F + f0 + j] = -acc;
                }
            }
        }
    }
}

extern "C" void kernel_launch(void* const* d_in, const int* in_sizes, int n_in,
                              void* d_out, int out_size, void* d_ws, size_t ws_size,
                              hipStream_t stream) {
    const float* x    = (const float*)d_in[0];  // [B, 5, F] f32
    const float* W    = (const float*)d_in[1];  // [F, 5]    f32
    const float* bias = (const float*)d_in[2];  // [F]       f32
    float* out = (float*)d_out;                 // [B, F]    f32

    const int F = in_sizes[2];                  // 25000
    const int B = in_sizes[0] / (WSZ * F);      // 512

    dim3 grid((F + FPB - 1) / FPB, (B + BPB - 1) / BPB);  // (25, 64)
    ocsvm_score_kernel<<<grid, TPB, 0, stream>>>(x, W, bias, out, B, F);
}